// ConditionalLayers_14851996910143
// MI455X (gfx1250) — compile-verified
//
#include <hip/hip_runtime.h>

typedef float v2f __attribute__((ext_vector_type(2)));
typedef float v8f __attribute__((ext_vector_type(8)));

// Problem constants (match reference)
constexpr int IN_CH  = 256;   // C (K dim)
constexpr int OUT_CH = 256;   // D (M dim)
constexpr int BATCH  = 64;
constexpr int TLEN   = 4096;  // T (N dim)

// Tiling: block = D 256 x T 128, 8 waves, each wave a 64x64 tile.
constexpr int BT    = 128;             // T tile per block
constexpr int BK    = 32;              // K chunk staged in LDS
constexpr int XROWF = BT * 2 + 32;     // 288 floats; 1152B % 256B == 128B -> k+2 reads
                                       //   land on the opposite bank half
constexpr int WROWF = OUT_CH * 2 + 32; // 544 floats; 2176B % 256B == 128B, same property
constexpr int NCHUNK = IN_CH / BK;     // 8

__global__ __launch_bounds__(256, 1)
void cond_gemm_f32_wmma(const float* __restrict__ x,
                        const float* __restrict__ w,
                        const int*   __restrict__ cond,
                        float*       __restrict__ out)
{
    // K-pair interleaved chunks: element (k,i) at buf[(k>>1)*ROWF + 2*i + (k&1)]
    __shared__ float xs [2][(BK / 2) * XROWF];  // 36 KB : x  chunk, i = t
    __shared__ float wsm[2][(BK / 2) * WROWF];  // 68 KB : W  chunk, i = d

    const int tid  = threadIdx.x;
    const int lane = tid & 31;
    const int wave = tid >> 5;

    const int b  = blockIdx.y;
    const int t0 = blockIdx.x * BT;

    const float* __restrict__ xb = x + (size_t)b * IN_CH * TLEN + t0;
    const float* __restrict__ wb = w + (size_t)cond[b] * IN_CH * OUT_CH;
    float*       __restrict__ ob = out + (size_t)b * OUT_CH * TLEN + t0;

    const int dBase = (wave & 3) * 64;   // 4 waves tile D
    const int tBase = (wave >> 2) * 64;  // 2 waves tile T

    // WMMA lane decomposition (ISA 7.12.2, 32-bit layouts)
    const int mrow  = lane & 15;   // M row (A) / N col (B,C)
    const int khalf = lane >> 4;   // 0: K,K+1 ; 1: K+2,K+3

    v8f acc[4][4];
#pragma unroll
    for (int i = 0; i < 4; ++i)
#pragma unroll
        for (int j = 0; j < 4; ++j)
            acc[i][j] = (v8f){0.f,0.f,0.f,0.f,0.f,0.f,0.f,0.f};

    // Stage a 32(K) x 128(T) x-chunk, K-pair interleaved. 512 units, 2/thread.
    auto stage_x = [&](int buf, int k0) {
#pragma unroll
        for (int j = 0; j < 2; ++j) {
            const int u  = tid + 256 * j;       // 0..511
            const int kp = u >> 5;              // 0..15 k-pair
            const int t4 = (u & 31) << 2;       // 0..124
            const float4 g0 =
                *reinterpret_cast<const float4*>(xb + (size_t)(k0 + 2 * kp)     * TLEN + t4);
            const float4 g1 =
                *reinterpret_cast<const float4*>(xb + (size_t)(k0 + 2 * kp + 1) * TLEN + t4);
            v2f* dst = reinterpret_cast<v2f*>(&xs[buf][kp * XROWF + t4 * 2]);
            dst[0] = (v2f){g0.x, g1.x};
            dst[1] = (v2f){g0.y, g1.y};
            dst[2] = (v2f){g0.z, g1.z};
            dst[3] = (v2f){g0.w, g1.w};
        }
    };

    // Stage a 32(K) x 256(D) W-chunk, K-pair interleaved. 1024 units, 4/thread.
    auto stage_w = [&](int buf, int k0) {
#pragma unroll
        for (int j = 0; j < 4; ++j) {
            const int u  = tid + 256 * j;       // 0..1023
            const int kp = u >> 6;              // 0..15 k-pair
            const int d4 = (u & 63) << 2;       // 0..252
            const float4 g0 =
                *reinterpret_cast<const float4*>(wb + (size_t)(k0 + 2 * kp)     * OUT_CH + d4);
            const float4 g1 =
                *reinterpret_cast<const float4*>(wb + (size_t)(k0 + 2 * kp + 1) * OUT_CH + d4);
            v2f* dst = reinterpret_cast<v2f*>(&wsm[buf][kp * WROWF + d4 * 2]);
            dst[0] = (v2f){g0.x, g1.x};
            dst[1] = (v2f){g0.y, g1.y};
            dst[2] = (v2f){g0.z, g1.z};
            dst[3] = (v2f){g0.w, g1.w};
        }
    };

    stage_x(0, 0);
    stage_w(0, 0);
    __syncthreads();

    for (int c = 0; c < NCHUNK; ++c) {
        const int k0 = c * BK;
        if (c + 1 < NCHUNK) {
            stage_x((c + 1) & 1, k0 + BK);
            stage_w((c + 1) & 1, k0 + BK);
        }
        const float* __restrict__ xbuf = xs [c & 1];
        const float* __restrict__ wbuf = wsm[c & 1];

#pragma unroll
        for (int kk = 0; kk < BK; kk += 4) {
            const int kb2 = (kk + 2 * khalf) >> 1;  // this lane-half's K-pair row

            // A fragments: {W[kb][d], W[kb+1][d]} adjacent in LDS -> one b64 load
            v2f afrag[4];
#pragma unroll
            for (int mh = 0; mh < 4; ++mh) {
                const int dloc = dBase + 16 * mh + mrow;
                afrag[mh] = *reinterpret_cast<const v2f*>(
                    &wbuf[kb2 * WROWF + dloc * 2]);
            }
            // B fragments: {x[kb][t], x[kb+1][t]} adjacent in LDS -> one b64 load
            v2f bfrag[4];
#pragma unroll
            for (int nq = 0; nq < 4; ++nq) {
                const int tloc = tBase + nq * 16 + mrow;
                bfrag[nq] = *reinterpret_cast<const v2f*>(
                    &xbuf[kb2 * XROWF + tloc * 2]);
            }
            // 16 WMMAs: D = A x B + C
#pragma unroll
            for (int mh = 0; mh < 4; ++mh)
#pragma unroll
                for (int nq = 0; nq < 4; ++nq)
                    acc[mh][nq] = __builtin_amdgcn_wmma_f32_16x16x4_f32(
                        false, afrag[mh], false, bfrag[nq],
                        (short)0, acc[mh][nq], false, false);
        }
        __syncthreads();
    }

    // Epilogue: C layout — VGPR r: lanes 0-15 -> M=r, lanes 16-31 -> M=r+8; N = mrow
    const int rAdd = (lane < 16) ? 0 : 8;
#pragma unroll
    for (int mh = 0; mh < 4; ++mh)
#pragma unroll
        for (int nq = 0; nq < 4; ++nq)
#pragma unroll
            for (int r = 0; r < 8; ++r) {
                const int d = dBase + 16 * mh + r + rAdd;
                ob[(size_t)d * TLEN + tBase + nq * 16 + mrow] = acc[mh][nq][r];
            }
}

extern "C" void kernel_launch(void* const* d_in, const int* in_sizes, int n_in,
                              void* d_out, int out_size, void* d_ws, size_t ws_size,
                              hipStream_t stream) {
    const float* x    = (const float*)d_in[0];
    const float* w    = (const float*)d_in[1];
    const int*   cond = (const int*)d_in[2];
    float* out = (float*)d_out;

    dim3 grid(TLEN / BT, BATCH);   // (32, 64)
    cond_gemm_f32_wmma<<<grid, 256, 0, stream>>>(x, w, cond, out);
}